// Decoding_33019708572164
// MI455X (gfx1250) — compile-verified
//
#include <hip/hip_runtime.h>
#include <math.h>

#define N_CELLS       1000
#define N_GENES_OI    500
#define N_GENES_TOTAL 5000
#define N_LATENT      10
#define NBINS         128
#define N_KNOTS       129   // NBINS + 1
#define N_CUTS        1000000

typedef float v2f __attribute__((ext_vector_type(2)));
typedef float v8f __attribute__((ext_vector_type(8)));

// ---------------------------------------------------------------------------
// Kernel 1: logits[c][g] = overall_baseline[g] + sum_l latent[c][l]*osw[g][l]
// One wave per 16x16 output tile, K=10 padded to 12 -> 3x v_wmma_f32_16x16x4_f32
// A (16x4 f32): lanes 0-15 hold M=lane, VGPR0=K, VGPR1=K+1; lanes 16-31 K+2/K+3
// B (4x16 f32): mirrored (row striped across lanes within a VGPR)
// C/D (16x16 f32): VGPR r -> M=r (lanes 0-15) / M=r+8 (lanes 16-31), N=lane&15
// ---------------------------------------------------------------------------
__global__ __launch_bounds__(128) void overall_logits_wmma(
    const float* __restrict__ latent, const float* __restrict__ osw,
    const float* __restrict__ baseline, float* __restrict__ logits)
{
  const int lane = threadIdx.x & 31;
  const int wave = __builtin_amdgcn_readfirstlane((int)(threadIdx.x >> 5));
  const int tilesN = (N_GENES_TOTAL + 15) / 16;   // 313
  const int tilesM = (N_CELLS + 15) / 16;         // 63
  int tile = blockIdx.x * 4 + wave;
  if (tile >= tilesM * tilesN) return;            // wave-uniform: EXEC stays full
  const int tm = tile / tilesN, tn = tile % tilesN;
  const int c0 = tm * 16, g0 = tn * 16;

  const int m  = lane & 15;
  const int kp = (lane >> 4) * 2;                 // 0 or 2 (K sub-pair per half-wave)
  int cc = c0 + m; if (cc > N_CELLS - 1)       cc = N_CELLS - 1;
  int gg = g0 + m; if (gg > N_GENES_TOTAL - 1) gg = N_GENES_TOTAL - 1;

  v8f acc = {};
#pragma unroll
  for (int s = 0; s < 3; ++s) {                   // K = 0..3, 4..7, 8..11 (10,11 zero)
    const int l0 = s * 4 + kp;
    v2f a, b;
    a.x = (l0     < N_LATENT) ? latent[cc * N_LATENT + l0]     : 0.f;
    a.y = (l0 + 1 < N_LATENT) ? latent[cc * N_LATENT + l0 + 1] : 0.f;
    b.x = (l0     < N_LATENT) ? osw[gg * N_LATENT + l0]        : 0.f;
    b.y = (l0 + 1 < N_LATENT) ? osw[gg * N_LATENT + l0 + 1]    : 0.f;
    acc = __builtin_amdgcn_wmma_f32_16x16x4_f32(false, a, false, b,
                                                (short)0, acc, false, false);
  }

  const int nO   = g0 + (lane & 15);
  const int half = (lane >> 4) * 8;
#pragma unroll
  for (int r = 0; r < 8; ++r) {
    const int cO = c0 + r + half;
    if (cO < N_CELLS && nO < N_GENES_TOTAL)
      logits[cO * N_GENES_TOTAL + nO] = acc[r] + baseline[nO];
  }
}

// ---------------------------------------------------------------------------
// Kernel 2: per-cell logsumexp over 5000 logits -> lse[c]
// ---------------------------------------------------------------------------
__global__ __launch_bounds__(256) void row_lse(
    const float* __restrict__ logits, float* __restrict__ lse)
{
  __shared__ float red[256];
  const int c = blockIdx.x;
  const float* row = logits + (size_t)c * N_GENES_TOTAL;

  float mx = -INFINITY;
  for (int g = threadIdx.x; g < N_GENES_TOTAL; g += 256) mx = fmaxf(mx, row[g]);
  red[threadIdx.x] = mx; __syncthreads();
  for (int s = 128; s > 0; s >>= 1) {
    if (threadIdx.x < s) red[threadIdx.x] = fmaxf(red[threadIdx.x], red[threadIdx.x + s]);
    __syncthreads();
  }
  mx = red[0]; __syncthreads();

  float ss = 0.f;
  for (int g = threadIdx.x; g < N_GENES_TOTAL; g += 256) ss += __expf(row[g] - mx);
  red[threadIdx.x] = ss; __syncthreads();
  for (int s = 128; s > 0; s >>= 1) {
    if (threadIdx.x < s) red[threadIdx.x] += red[threadIdx.x + s];
    __syncthreads();
  }
  if (threadIdx.x == 0) lse[c] = mx + __logf(red[0]);
}

// ---------------------------------------------------------------------------
// Kernel 3: one wave32 per cut. Lanes split the 129 knots (k = lane + 32*j).
// h[k] = spline_baseline[g][k] + sum_l latent[c][l] * hsw[g][l][k]
// Stable spline log-prob + log(overall) + log(5000); block-reduced atomicAdd.
// ---------------------------------------------------------------------------
__global__ __launch_bounds__(256) void cuts_kernel(
    const float* __restrict__ latent,
    const float* __restrict__ cut_coord,
    const int*   __restrict__ genes_oi,
    const int*   __restrict__ cut_cg,    // c*500 + goi
    const int*   __restrict__ cut_cG,    // c*5000 + g
    const int*   __restrict__ cut_goi,   // goi
    const float* __restrict__ hsw,       // [5000][10][129]
    const float* __restrict__ sb,        // [5000][129]
    const float* __restrict__ logits,
    const float* __restrict__ lse,
    float*       __restrict__ out)
{
  __shared__ float wsum[8];
  const int lane = threadIdx.x & 31;
  const int wave = __builtin_amdgcn_readfirstlane((int)(threadIdx.x >> 5));
  const int cut  = blockIdx.x * 8 + wave;

  float contrib = 0.f;
  if (cut < N_CUTS) {
    const int   ixcg = cut_cg[cut];              // uniform (scalar loads)
    const int   c    = ixcg / N_GENES_OI;
    const int   g    = genes_oi[cut_goi[cut]];
    const int   ixCG = cut_cG[cut];
    const float x    = cut_coord[cut];

    float lat[N_LATENT];
#pragma unroll
    for (int l = 0; l < N_LATENT; ++l) lat[l] = latent[c * N_LATENT + l];

    float h[5];
#pragma unroll
    for (int j = 0; j < 5; ++j) {
      const int k = lane + 32 * j;
      if (k < N_KNOTS) {
        const float* wp = hsw + (size_t)g * (N_LATENT * N_KNOTS) + k;
        float a = sb[g * N_KNOTS + k];
#pragma unroll
        for (int l = 0; l < N_LATENT; ++l) a = fmaf(lat[l], wp[l * N_KNOTS], a);
        h[j] = a;
      } else {
        h[j] = -INFINITY;                        // exp -> 0, excluded naturally
      }
    }

    float mx = fmaxf(fmaxf(fmaxf(h[0], h[1]), fmaxf(h[2], h[3])), h[4]);
#pragma unroll
    for (int o = 16; o > 0; o >>= 1) mx = fmaxf(mx, __shfl_xor(mx, o, 32));

    float u[5], ssum = 0.f;
#pragma unroll
    for (int j = 0; j < 5; ++j) { u[j] = __expf(h[j] - mx); ssum += u[j]; }
#pragma unroll
    for (int o = 16; o > 0; o >>= 1) ssum += __shfl_xor(ssum, o, 32);

    // trapezoid integral = (sum u - 0.5*u[0] - 0.5*u[128]) / NBINS
    const float u_first = __shfl(u[0], 0);       // k = 0   lives in lane 0 slot 0
    const float u_last  = __shfl(u[4], 0);       // k = 128 lives in lane 0 slot 4
    const float norm = (ssum - 0.5f * u_first - 0.5f * u_last) * (1.0f / NBINS);

    float xs = fminf(fmaxf(x, 0.f), 1.f - 1e-6f) * (float)NBINS;
    int b = (int)floorf(xs);
    b = b < 0 ? 0 : (b > NBINS - 1 ? NBINS - 1 : b);
    const float alpha = xs - (float)b;

    const int s0 = b >> 5;                        // uniform per wave
    const float cand0 = s0 == 0 ? u[0] : s0 == 1 ? u[1] : s0 == 2 ? u[2]
                                : s0 == 3 ? u[3] : u[4];
    const float ub  = __shfl(cand0, b & 31);
    const int s1 = (b + 1) >> 5;
    const float cand1 = s1 == 0 ? u[0] : s1 == 1 ? u[1] : s1 == 2 ? u[2]
                                : s1 == 3 ? u[3] : u[4];
    const float ub1 = __shfl(cand1, (b + 1) & 31);

    const float val    = ub + alpha * (ub1 - ub);
    const float loglik = __logf(val) - __logf(norm);
    const float log_overall = logits[ixCG] - lse[ixCG / N_GENES_TOTAL];
    contrib = -(loglik + log_overall + 8.51719319141623742f /* log 5000 */);
  }

  if (lane == 0) wsum[wave] = contrib;
  __syncthreads();
  if (threadIdx.x == 0) {
    float t = 0.f;
#pragma unroll
    for (int w = 0; w < 8; ++w) t += wsum[w];
    atomicAdd(out, t);
  }
}

// ---------------------------------------------------------------------------
extern "C" void kernel_launch(void* const* d_in, const int* in_sizes, int n_in,
                              void* d_out, int out_size, void* d_ws, size_t ws_size,
                              hipStream_t stream)
{
  const float* latent   = (const float*)d_in[0];
  const float* cutc     = (const float*)d_in[1];
  const int*   genes_oi = (const int*)  d_in[2];
  const int*   cut_cg   = (const int*)  d_in[3];
  const int*   cut_cG   = (const int*)  d_in[4];
  const int*   cut_goi  = (const int*)  d_in[5];
  const float* hsw      = (const float*)d_in[6];
  const float* osw      = (const float*)d_in[7];
  const float* obase    = (const float*)d_in[8];
  const float* sb       = (const float*)d_in[9];
  float* out = (float*)d_out;

  float* logits = (float*)d_ws;                                  // 20,000,000 B
  float* lse    = logits + (size_t)N_CELLS * N_GENES_TOTAL;      // + 4,000 B

  hipMemsetAsync(out, 0, sizeof(float), stream);                 // graph-safe

  const int tiles = ((N_CELLS + 15) / 16) * ((N_GENES_TOTAL + 15) / 16); // 19719
  overall_logits_wmma<<<(tiles + 3) / 4, 128, 0, stream>>>(latent, osw, obase, logits);
  row_lse<<<N_CELLS, 256, 0, stream>>>(logits, lse);
  cuts_kernel<<<(N_CUTS + 7) / 8, 256, 0, stream>>>(latent, cutc, genes_oi,
      cut_cg, cut_cG, cut_goi, hsw, sb, logits, lse, out);
}